// RVQ_56538949484662
// MI455X (gfx1250) — compile-verified
//
#include <hip/hip_runtime.h>

#define H_   3
#define K16  16
#define D_   768
#define DH_  256
#define RPAD 260                    // padded codebook row stride (floats): 260 % 64 = 4 -> conflict-free b64 reads
#define TOK_PER_WAVE 16
#define WAVES_PER_BLOCK 8
#define TOK_PER_BLOCK (TOK_PER_WAVE * WAVES_PER_BLOCK)
#define NTOK (16 * 4096)

typedef __attribute__((ext_vector_type(2))) float v2f;
typedef __attribute__((ext_vector_type(8))) float v8f;
typedef __attribute__((ext_vector_type(4))) float f4;   // native vector (builtin-compatible)
typedef int v4i __attribute__((vector_size(16)));       // async-builtin arg0 pointee
typedef __attribute__((address_space(3))) v4i v4i_lds;  // async-builtin arg1 pointee (AS3)

#if __has_builtin(__builtin_amdgcn_global_load_async_to_lds_b128)
#define HAVE_ASYNC_LDS 1
#endif

static __device__ __forceinline__ void wait_asynccnt0() {
#if __has_builtin(__builtin_amdgcn_s_wait_asynccnt)
    __builtin_amdgcn_s_wait_asynccnt(0);
#else
    asm volatile("s_wait_asynccnt 0" ::: "memory");
#endif
}

#if __has_builtin(__builtin_amdgcn_update_dpp)
#define HAVE_DPP 1
template <int CTRL>
static __device__ __forceinline__ float dppmin(float v) {
    int s = __builtin_bit_cast(int, v);
    int p = __builtin_amdgcn_update_dpp(s, s, CTRL, 0xF, 0xF, true);
    return fminf(v, __builtin_bit_cast(float, p));
}
#endif

// min over each independent 16-lane row (lanes 0-15 and 16-31), pure VALU via DPP16
static __device__ __forceinline__ float rowmin16(float v) {
#ifdef HAVE_DPP
    v = dppmin<0xB1>(v);    // quad_perm(1,0,3,2) : xor 1
    v = dppmin<0x4E>(v);    // quad_perm(2,3,0,1) : xor 2
    v = dppmin<0x141>(v);   // row_half_mirror    : xor 7
    v = dppmin<0x140>(v);   // row_mirror         : xor 15
#else
    #pragma unroll
    for (int m = 1; m < 16; m <<= 1)
        v = fminf(v, __shfl_xor(v, m, 32));
#endif
    return v;
}

__global__ __launch_bounds__(256) void rvq_wmma_kernel(
    const float* __restrict__ x,       // [NTOK, 768]
    const float* __restrict__ cb,      // [3, 16, 256]
    const float* __restrict__ alpha_p, // scalar
    float* __restrict__ out,           // [NTOK, 768]
    float* __restrict__ code)          // [NTOK] (codes stored as float)
{
    __shared__ float lds_cb[H_ * K16 * RPAD];   // 12480 floats, padded rows
    __shared__ float lds_knorm[H_ * K16];       // 48
    __shared__ int   lds_ind[TOK_PER_BLOCK * H_];

    const int tid     = threadIdx.x;
    const int lane    = tid & 31;
    const int wave    = tid >> 5;
    const int halfsel = (lane >> 4) & 1;        // 0: lanes 0-15, 1: lanes 16-31
    const int l15     = lane & 15;

    // ---- Phase 0: stage codebook into LDS (padded rows) ----
    // 12288 floats = 3072 x 16B over 256 threads = 12 iters
    #pragma unroll
    for (int i = 0; i < 12; ++i) {
        int vid = i * 256 + tid;                // 16B-chunk id
        int g   = vid * 4;                      // source float offset
        int cwg = g / DH_;                      // 0..47 (h*16 + cw)
        int dd  = g - cwg * DH_;                // 0..252, multiple of 4
#ifdef HAVE_ASYNC_LDS
        // gfx1250 async VMEM->LDS (ASYNCcnt-tracked), per-lane addresses
        __builtin_amdgcn_global_load_async_to_lds_b128(
            (v4i*)(cb + g), (v4i_lds*)(lds_cb + cwg * RPAD + dd), 0, 0);
#else
        f4 val = *(const f4*)(cb + g);
        *(f4*)(lds_cb + cwg * RPAD + dd) = val;
#endif
    }
#ifdef HAVE_ASYNC_LDS
    wait_asynccnt0();
#endif
    __syncthreads();

    // ---- codebook squared norms ----
    if (tid < H_ * K16) {
        const float* row = lds_cb + tid * RPAD;
        float s = 0.0f;
        #pragma unroll 8
        for (int d = 0; d < DH_; ++d) s += row[d] * row[d];
        lds_knorm[tid] = s;
    }
    __syncthreads();

    // ---- Phase 1: per-wave 16-token tile, WMMA f32 scores + argmin ----
    const int t0   = (blockIdx.x * WAVES_PER_BLOCK + wave) * TOK_PER_WAVE;
    const int wtb  = wave * TOK_PER_WAVE;
    const int tokA = t0 + l15;                  // A-matrix row (M = lane & 15)
    // f32 A layout (16x4): VGPR0 = K{0 | 2}, VGPR1 = K{1 | 3}; halfsel picks +2
    const float* xrowA = x + (size_t)tokA * D_ + halfsel * 2;

    for (int h = 0; h < H_; ++h) {
        v8f acc = {};                           // C/D: VGPR j -> rows {j, j+8}, N = lane&15
        const float* ap = xrowA + h * DH_;
        // B layout (4x16) mirrors A with M->N: VGPR0 = K{0|2}, VGPR1 = K{1|3}, N = lane&15
        const float* bp = lds_cb + (h * K16 + l15) * RPAD + halfsel * 2;

        if (h + 1 < H_) __builtin_prefetch(ap + DH_, 0, 3);  // WGP-scope global_prefetch_b8

        #pragma unroll 8
        for (int k = 0; k < DH_; k += 4) {
            v2f a = *(const v2f*)(ap + k);      // global_load_b64
            v2f b = *(const v2f*)(bp + k);      // ds_load_b64 (conflict-free via RPAD)
            acc = __builtin_amdgcn_wmma_f32_16x16x4_f32(
                false, a, false, b, (short)0, acc, false, false);
        }

        // argmin over n of (||k_n||^2 - 2*dot[m][n]); row-constant ||x||^2 dropped.
        // Branch-free: 4-step DPP16 min reduction (16-lane rows independent),
        // then ballot + ffs picks the FIRST minimal index (jnp.argmin tie-break).
        const float kn = lds_knorm[h * K16 + l15];
        #pragma unroll
        for (int j = 0; j < 8; ++j) {
            float v  = kn - 2.0f * acc[j];
            float vm = rowmin16(v);
#if __has_builtin(__builtin_amdgcn_ballot_w32)
            unsigned bal = (unsigned)__builtin_amdgcn_ballot_w32(v == vm);
#else
            unsigned bal = (unsigned)__ballot(v == vm);
#endif
            if (l15 == 0) {
                unsigned half = halfsel ? (bal >> 16) : (bal & 0xffffu);
                int row = j + halfsel * 8;      // token row within tile
                lds_ind[(wtb + row) * H_ + h] = __ffs(half) - 1;
            }
        }
    }
    __syncthreads();

    // ---- Phase 2: convex blend + code pack ----
    const float alpha = *alpha_p;
    const float beta  = 1.0f - alpha;

    // 16 tokens * 768 floats = 3072 x f4 over 32 lanes = 96 iters
    #pragma unroll 4
    for (int i = 0; i < 96; ++i) {
        int vid  = i * 32 + lane;               // f4 id within tile, 0..3071
        int m    = vid / 192;                   // 192 f4 per token
        int q    = vid - m * 192;
        int dcol = q * 4;                       // float column, mult of 4
        int h    = dcol >> 8;
        int dd   = dcol & 255;
        int tok  = t0 + m;

        f4 xv = *(const f4*)(x + (size_t)tok * D_ + dcol);  // hot in WGP$/L2
        int ind = lds_ind[(wtb + m) * H_ + h];
        f4 kv = *(const f4*)(lds_cb + (h * K16 + ind) * RPAD + dd);

        f4 o;
        o.x = alpha * xv.x + beta * kv.x;
        o.y = alpha * xv.y + beta * kv.y;
        o.z = alpha * xv.z + beta * kv.z;
        o.w = alpha * xv.w + beta * kv.w;
        // streaming write-only output: non-temporal so it doesn't evict x/cb from L2
        __builtin_nontemporal_store(o, (f4*)(out + (size_t)tok * D_ + dcol));
    }

    if (lane < 16) {
        int tok = t0 + lane;
        int i0 = lds_ind[(wtb + lane) * H_ + 0];
        int i1 = lds_ind[(wtb + lane) * H_ + 1];
        int i2 = lds_ind[(wtb + lane) * H_ + 2];
        code[tok] = (float)(i0 + 16 * i1 + 256 * i2);
    }
}

extern "C" void kernel_launch(void* const* d_in, const int* in_sizes, int n_in,
                              void* d_out, int out_size, void* d_ws, size_t ws_size,
                              hipStream_t stream) {
    (void)in_sizes; (void)n_in; (void)d_ws; (void)ws_size; (void)out_size;
    const float* x     = (const float*)d_in[0];
    const float* cb    = (const float*)d_in[1];
    const float* alpha = (const float*)d_in[2];
    float* out  = (float*)d_out;
    float* code = (float*)d_out + (size_t)NTOK * D_;   // outputs concatenated flat

    dim3 grid(NTOK / TOK_PER_BLOCK);   // 512
    dim3 block(256);                   // 8 wave32s
    rvq_wmma_kernel<<<grid, block, 0, stream>>>(x, cb, alpha, out, code);
}